// ChamferDistanceL1AndSeg_78383153152283
// MI455X (gfx1250) — compile-verified
//
#include <hip/hip_runtime.h>

typedef float v2f __attribute__((ext_vector_type(2)));
typedef float v8f __attribute__((ext_vector_type(8)));

#define BATCH 8
#define NPTS 4096
#define TILES (NPTS / 16)   // 256 row tiles per batch, 256 column tiles

// One wave (32 lanes) owns a 16-row tile of A for one batch and scans all 4096
// columns of B in 16-wide tiles. Each tile is one V_WMMA_F32_16X16X4_F32 with
// the spare K=3 slot carrying ||y||^2:
//   A = [-2x, -2y | -2z, 1],  B = [bx, by | bz, ||b||^2],  C = 0
//   => D[m,n] = ||b_n||^2 - 2<a_m, b_n>
// ||a_m||^2 is row-constant, so it is added after the min-scan (commutes with
// min), and the >=0 clamp is applied once at the end (max(.,0) is monotone).
__global__ __launch_bounds__(256) void chamfer_pass_kernel(
    const float* __restrict__ xyzA, const float* __restrict__ xyzB,
    int* __restrict__ idx_out, float* __restrict__ wave_sums)
{
  const int lane  = threadIdx.x & 31;
  const int half  = lane >> 4;       // selects K-pair {0,1} vs {2,3}
  const int m16   = lane & 15;       // row index (A frag) / col index (B frag)
  const int waveg = (int)((blockIdx.x * blockDim.x + threadIdx.x) >> 5);
  const int b        = waveg >> 8;               // 256 row tiles per batch
  const int row_base = (waveg & (TILES - 1)) << 4;

  const float* Ab = xyzA + (size_t)b * NPTS * 3;
  const float* Bb = xyzB + (size_t)b * NPTS * 3;

  // --- A fragment (fixed for this wave); K=3 slot = 1.0 picks up ||b||^2 ---
  const int arow = row_base + m16;
  const float ax = Ab[arow * 3 + 0];
  const float ay = Ab[arow * 3 + 1];
  const float az = Ab[arow * 3 + 2];
  const float sqa = ax * ax + ay * ay + az * az;

  v2f afrag;
  afrag.x = -2.0f * (half ? az : ax);
  afrag.y = half ? 1.0f : (-2.0f * ay);

  float vmin[8];
  int   vidx[8];
#pragma unroll
  for (int r = 0; r < 8; ++r) { vmin[r] = 3.4e38f; vidx[r] = 0; }

  // software-pipelined B-point load (one tile ahead)
  float bx = Bb[m16 * 3 + 0];
  float by = Bb[m16 * 3 + 1];
  float bz = Bb[m16 * 3 + 2];

#pragma unroll 2
  for (int ct = 0; ct < TILES; ++ct) {
    const int coln = ((((ct + 1) & (TILES - 1)) << 4) + m16) * 3;
    const float nx = Bb[coln + 0];              // prefetch next tile's point
    const float ny = Bb[coln + 1];
    const float nz = Bb[coln + 2];

    const float sqb = bx * bx + by * by + bz * bz;
    v2f bfrag;                                  // K=3 slot carries ||b||^2
    bfrag.x = half ? bz : bx;
    bfrag.y = half ? sqb : by;

    v8f c = {};                                 // inline-0 accumulator
    v8f d = __builtin_amdgcn_wmma_f32_16x16x4_f32(
        false, afrag, false, bfrag, (short)0, c, false, false);

    const int col = (ct << 4) + m16;
#pragma unroll
    for (int r = 0; r < 8; ++r) {
      if (d[r] < vmin[r]) { vmin[r] = d[r]; vidx[r] = col; }  // first-occurrence
    }

    bx = nx; by = ny; bz = nz;
  }

  // --- min+argmin butterfly within each 16-lane half (column classes) ---
#pragma unroll
  for (int off = 1; off <= 8; off <<= 1) {
#pragma unroll
    for (int r = 0; r < 8; ++r) {
      float om = __shfl_xor(vmin[r], off, 32);
      int   oi = __shfl_xor(vidx[r], off, 32);
      if (om < vmin[r] || (om == vmin[r] && oi < vidx[r])) {
        vmin[r] = om; vidx[r] = oi;
      }
    }
  }

  // lanes with m16==0 hold rows [row_base+8*half .. +7]
  if (idx_out != nullptr && m16 == 0) {
#pragma unroll
    for (int r = 0; r < 8; ++r)
      idx_out[(size_t)b * NPTS + row_base + 8 * half + r] = vidx[r];
  }

  // add row-constant ||a||^2, clamp, sqrt, per-wave partial sum (fixed order)
  float s = 0.0f;
#pragma unroll
  for (int r = 0; r < 8; ++r) {
    const float sqa_r = __shfl(sqa, r + 8 * half, 32);
    s += sqrtf(fmaxf(vmin[r] + sqa_r, 0.0f));
  }
  s += __shfl_xor(s, 16, 32);                   // combine both halves
  if (lane == 0) wave_sums[waveg] = s;
}

// Deterministic single-block tree reduction of the 4096 wave partials.
__global__ __launch_bounds__(256) void chamfer_reduce_kernel(
    const float* __restrict__ partial, int n, float* __restrict__ out,
    float scale)
{
  __shared__ float sh[256];
  float s = 0.0f;
  for (int i = threadIdx.x; i < n; i += 256) s += partial[i];
  sh[threadIdx.x] = s;
  __syncthreads();
  for (int off = 128; off > 0; off >>= 1) {
    if ((int)threadIdx.x < off) sh[threadIdx.x] += sh[threadIdx.x + off];
    __syncthreads();
  }
  if (threadIdx.x == 0) out[0] = sh[0] * scale;
}

extern "C" void kernel_launch(void* const* d_in, const int* in_sizes, int n_in,
                              void* d_out, int out_size, void* d_ws, size_t ws_size,
                              hipStream_t stream) {
  (void)in_sizes; (void)n_in; (void)out_size; (void)ws_size;
  const float* xyz1 = (const float*)d_in[0];   // [8,4096,3] f32
  const float* xyz2 = (const float*)d_in[1];   // [8,4096,3] f32
  float* loss_out = (float*)d_out;             // element 0: cd_loss
  int*   idx_out  = ((int*)d_out) + 1;         // elements 1..32768: idx1
  float* wsum     = (float*)d_ws;              // 4096 wave partial sums

  const int waves_per_pass = BATCH * TILES;    // 2048
  const int blocks = waves_per_pass / 8;       // 256 blocks x 8 waves

  // Pass 1: for each xyz1 point, nearest in xyz2 (+ indices)
  chamfer_pass_kernel<<<blocks, 256, 0, stream>>>(xyz1, xyz2, idx_out, wsum);
  // Pass 2: for each xyz2 point, nearest in xyz1
  chamfer_pass_kernel<<<blocks, 256, 0, stream>>>(xyz2, xyz1, nullptr,
                                                  wsum + waves_per_pass);
  // loss = 0.5*(sum1 + sum2)/32768
  chamfer_reduce_kernel<<<1, 256, 0, stream>>>(wsum, 2 * waves_per_pass,
                                               loss_out, 0.5f / 32768.0f);
}